// GAT_6975026888991
// MI455X (gfx1250) — compile-verified
//
#include <hip/hip_runtime.h>
#include <hip/hip_bf16.h>
#include <math.h>

#define N_NODES 50000
#define N_EDGES 800000
#define HEADS   4
#define SLOPE   0.2f
#define EPSBN   1e-5f

typedef float v2f __attribute__((ext_vector_type(2)));
typedef float v8f __attribute__((ext_vector_type(8)));

// ---------------------------------------------------------------- utilities
__global__ void fill_kernel(float* __restrict__ p, float v, int n) {
    int i = blockIdx.x * blockDim.x + threadIdx.x;
    if (i < n) p[i] = v;
}

__device__ inline void atomicMaxF(float* addr, float val) {
    // works given buffer initialized to -inf
    if (val >= 0.f) atomicMax((int*)addr, __float_as_int(val));
    else            atomicMin((unsigned int*)addr, __float_as_uint(val));
}

// ---------------------------------------------------------------- fp32 WMMA GEMM
// Out[M=rows x NCOLS] = A[rows x 128] * W[128 x NCOLS]
// One wave -> one 16x16 tile; K-loop of 32 x V_WMMA_F32_16X16X4_F32.
// NCOLS is a template constant (128 or 160) so all W offsets are immediates.
template<int NCOLS>
__global__ __launch_bounds__(256)
void gemm_f32_wmma(const float* __restrict__ A, const float* __restrict__ W,
                   float* __restrict__ Out) {
    __shared__ float As[16][132];            // 132 pad: conflict-free, float4-aligned
    const int tid  = threadIdx.x;
    const int row0 = blockIdx.x * 16;

    // cooperative 16x128 A-tile load (512 float4 slots / 256 threads)
    #pragma unroll
    for (int i = 0; i < 2; ++i) {
        int idx = tid + i * 256;
        int r   = idx >> 5;                  // idx / 32
        int c4  = idx & 31;
        float4 v = *(const float4*)(A + (size_t)(row0 + r) * 128 + c4 * 4);
        *(float4*)(&As[r][c4 * 4]) = v;
    }
    __syncthreads();

    const int wave    = tid >> 5;
    const int lane    = tid & 31;
    const int colTile = blockIdx.y * 8 + wave;
    if (colTile * 16 >= NCOLS) return;       // wave-uniform exit (EXEC stays all-1s)
    const int col0 = colTile * 16;
    const int half = lane >> 4;              // 0: lanes 0-15, 1: lanes 16-31
    const int l16  = lane & 15;

    // hoisted B base: per-lane column + this half's K phase (2*half rows)
    const float* __restrict__ Wp = W + (size_t)(2 * half) * NCOLS + col0 + l16;

    v8f acc = {};
    #pragma unroll
    for (int kk = 0; kk < 128; kk += 4) {
        v2f a; a.x = As[l16][kk + 2 * half];
               a.y = As[l16][kk + 2 * half + 1];
        // W offsets are compile-time immediates after unrolling
        v2f b; b.x = Wp[kk * NCOLS];
               b.y = Wp[kk * NCOLS + NCOLS];
        acc = __builtin_amdgcn_wmma_f32_16x16x4_f32(
                  false, a, false, b, (short)0, acc, false, false);
    }
    float* __restrict__ Op = Out + (size_t)(row0 + half * 8) * NCOLS + col0 + l16;
    #pragma unroll
    for (int r = 0; r < 8; ++r)              // D layout: VGPR r -> M = half*8 + r
        Op[r * NCOLS] = acc[r];
}

// ---------------------------------------------------------------- attention logits
__global__ void eler_kernel(const float* __restrict__ z, const float* __restrict__ al,
                            const float* __restrict__ ar, float* __restrict__ el,
                            float* __restrict__ er, int fout) {
    int t = blockIdx.x * blockDim.x + threadIdx.x;
    if (t >= N_NODES * HEADS) return;
    int n = t / HEADS, h = t % HEADS;
    const float* zp  = z  + (size_t)n * HEADS * fout + h * fout;
    const float* alp = al + h * fout;
    const float* arp = ar + h * fout;
    float sl = 0.f, sr = 0.f;
    for (int d = 0; d < fout; ++d) { float zv = zp[d]; sl += zv * alp[d]; sr += zv * arp[d]; }
    el[t] = sl; er[t] = sr;
}

// ---------------------------------------------------------------- edge softmax
__global__ void edge_max_kernel(const int* __restrict__ src, const int* __restrict__ dst,
                                const float* __restrict__ el, const float* __restrict__ er,
                                float* __restrict__ ebuf, float* __restrict__ m) {
    int t = blockIdx.x * blockDim.x + threadIdx.x;
    if (t >= N_EDGES * HEADS) return;
    int e = t / HEADS, h = t % HEADS;
    int s = src[e], d = dst[e];
    float v = el[s * HEADS + h] + er[d * HEADS + h];
    v = (v > 0.f) ? v : SLOPE * v;           // leaky_relu
    ebuf[t] = v;
    atomicMaxF(&m[d * HEADS + h], v);
}

__global__ void edge_exp_kernel(const int* __restrict__ dst, float* __restrict__ ebuf,
                                const float* __restrict__ m, float* __restrict__ sden) {
    int t = blockIdx.x * blockDim.x + threadIdx.x;
    if (t >= N_EDGES * HEADS) return;
    int e = t / HEADS, h = t % HEADS;
    int d = dst[e];
    float ex = __expf(ebuf[t] - m[d * HEADS + h]);
    ebuf[t] = ex;
    atomicAdd(&sden[d * HEADS + h], ex);
}

// ---------------------------------------------------------------- weighted scatter-add
// one wave per edge; lane covers PER consecutive channels (all in one head)
template<int FOUT, int PER>
__global__ __launch_bounds__(256)
void edge_agg_kernel(const int* __restrict__ src, const int* __restrict__ dst,
                     const float* __restrict__ z, const float* __restrict__ ebuf,
                     const float* __restrict__ sden, float* __restrict__ agg) {
    const int wave = threadIdx.x >> 5, lane = threadIdx.x & 31;
    const int e = blockIdx.x * 8 + wave;
    if (e >= N_EDGES) return;
    const int s = src[e], d = dst[e];
    const int base = lane * PER;
    const int h = base / FOUT;
    float sv    = sden[d * HEADS + h];
    float alpha = ebuf[e * HEADS + h] / (sv == 0.f ? 1.f : sv);
    const float* zp = z   + (size_t)s * (HEADS * FOUT) + base;
    float*       ap = agg + (size_t)d * (HEADS * FOUT) + base;
    #pragma unroll
    for (int j = 0; j < PER; ++j) atomicAdd(&ap[j], alpha * zp[j]);
}

// ---------------------------------------------------------------- epilogue pieces
__global__ void combine_kernel(float* __restrict__ agg, const float* __restrict__ zl,
                               const float* __restrict__ bc, int ncols, int total) {
    int t = blockIdx.x * blockDim.x + threadIdx.x;
    if (t >= total) return;
    agg[t] += zl[t] + bc[t % ncols];
}

__global__ __launch_bounds__(256)
void bn_stats_kernel(const float* __restrict__ x, float* __restrict__ mean,
                     float* __restrict__ var) {
    const int c = blockIdx.x;                // 0..127
    float s = 0.f, s2 = 0.f;
    for (int n = threadIdx.x; n < N_NODES; n += 256) {
        float v = x[(size_t)n * 128 + c];
        s += v; s2 += v * v;
    }
    __shared__ float sh[256], sh2[256];
    sh[threadIdx.x] = s; sh2[threadIdx.x] = s2;
    __syncthreads();
    for (int st = 128; st > 0; st >>= 1) {
        if (threadIdx.x < st) {
            sh[threadIdx.x]  += sh[threadIdx.x + st];
            sh2[threadIdx.x] += sh2[threadIdx.x + st];
        }
        __syncthreads();
    }
    if (threadIdx.x == 0) {
        float mu = sh[0] / (float)N_NODES;
        mean[c] = mu;
        var[c]  = sh2[0] / (float)N_NODES - mu * mu;
    }
}

__global__ void bn_relu_kernel(const float* __restrict__ x, const float* __restrict__ mean,
                               const float* __restrict__ var, const float* __restrict__ g,
                               const float* __restrict__ b, float* __restrict__ y) {
    int t = blockIdx.x * blockDim.x + threadIdx.x;
    if (t >= N_NODES * 128) return;
    int c = t & 127;
    float v = (x[t] - mean[c]) * rsqrtf(var[c] + EPSBN) * g[c] + b[c];
    y[t] = v > 0.f ? v : 0.f;
}

__global__ void head_mean_kernel(const float* __restrict__ x, const float* __restrict__ bias,
                                 float* __restrict__ out) {
    const int C = 40;
    int t = blockIdx.x * blockDim.x + threadIdx.x;
    if (t >= N_NODES * C) return;
    int n = t / C, c = t % C;
    const float* xp = x + (size_t)n * HEADS * C;
    float s = 0.f;
    #pragma unroll
    for (int h = 0; h < HEADS; ++h) s += xp[h * C + c];
    out[t] = s * (1.f / HEADS) + bias[c];
}

// ---------------------------------------------------------------- driver
extern "C" void kernel_launch(void* const* d_in, const int* in_sizes, int n_in,
                              void* d_out, int out_size, void* d_ws, size_t ws_size,
                              hipStream_t stream) {
    (void)in_sizes; (void)n_in; (void)out_size; (void)ws_size;
    const float* feat = (const float*)d_in[0];
    const int*   src  = (const int*)d_in[1];
    const int*   dst  = (const int*)d_in[2];
    const float* Wc[3] = {(const float*)d_in[3],  (const float*)d_in[8],  (const float*)d_in[13]};
    const float* al[3] = {(const float*)d_in[4],  (const float*)d_in[9],  (const float*)d_in[14]};
    const float* ar[3] = {(const float*)d_in[5],  (const float*)d_in[10], (const float*)d_in[15]};
    const float* bc[3] = {(const float*)d_in[6],  (const float*)d_in[11], (const float*)d_in[16]};
    const float* Wl[3] = {(const float*)d_in[7],  (const float*)d_in[12], (const float*)d_in[17]};
    const float* g[2]  = {(const float*)d_in[18], (const float*)d_in[20]};
    const float* bb[2] = {(const float*)d_in[19], (const float*)d_in[21]};
    const float* bias_last = (const float*)d_in[22];

    float* ws      = (float*)d_ws;
    float* bufX    = ws;                                   // N*128
    float* bufZc   = bufX   + (size_t)N_NODES * 128;       // N*160
    float* bufZl   = bufZc  + (size_t)N_NODES * 160;       // N*160
    float* bufAgg  = bufZl  + (size_t)N_NODES * 160;       // N*160
    float* bufE    = bufAgg + (size_t)N_NODES * 160;       // E*H
    float* bufM    = bufE   + (size_t)N_EDGES * HEADS;     // N*H
    float* bufS    = bufM   + (size_t)N_NODES * HEADS;     // N*H
    float* bufEl   = bufS   + (size_t)N_NODES * HEADS;     // N*H
    float* bufEr   = bufEl  + (size_t)N_NODES * HEADS;     // N*H
    float* bufMean = bufEr  + (size_t)N_NODES * HEADS;     // 128
    float* bufVar  = bufMean + 128;                        // 128

    const int fouts[3] = {32, 32, 40};
    const float* x = feat;
    const int nh = N_NODES * HEADS;
    const int eh = N_EDGES * HEADS;

    for (int L = 0; L < 3; ++L) {
        const int fout  = fouts[L];
        const int ncols = HEADS * fout;               // 128 or 160
        const int tot   = N_NODES * ncols;

        dim3 gg(N_NODES / 16, (ncols + 127) / 128);
        if (ncols == 128) {
            gemm_f32_wmma<128><<<gg, 256, 0, stream>>>(x, Wc[L], bufZc);
            gemm_f32_wmma<128><<<gg, 256, 0, stream>>>(x, Wl[L], bufZl);
        } else {
            gemm_f32_wmma<160><<<gg, 256, 0, stream>>>(x, Wc[L], bufZc);
            gemm_f32_wmma<160><<<gg, 256, 0, stream>>>(x, Wl[L], bufZl);
        }

        eler_kernel<<<(nh + 255) / 256, 256, 0, stream>>>(bufZc, al[L], ar[L], bufEl, bufEr, fout);

        fill_kernel<<<(nh + 255) / 256, 256, 0, stream>>>(bufM, -INFINITY, nh);
        fill_kernel<<<(nh + 255) / 256, 256, 0, stream>>>(bufS, 0.f, nh);
        fill_kernel<<<(tot + 255) / 256, 256, 0, stream>>>(bufAgg, 0.f, tot);

        edge_max_kernel<<<(eh + 255) / 256, 256, 0, stream>>>(src, dst, bufEl, bufEr, bufE, bufM);
        edge_exp_kernel<<<(eh + 255) / 256, 256, 0, stream>>>(dst, bufE, bufM, bufS);

        if (fout == 32)
            edge_agg_kernel<32, 4><<<N_EDGES / 8, 256, 0, stream>>>(src, dst, bufZc, bufE, bufS, bufAgg);
        else
            edge_agg_kernel<40, 5><<<N_EDGES / 8, 256, 0, stream>>>(src, dst, bufZc, bufE, bufS, bufAgg);

        combine_kernel<<<(tot + 255) / 256, 256, 0, stream>>>(bufAgg, bufZl, bc[L], ncols, tot);

        if (L < 2) {
            bn_stats_kernel<<<128, 256, 0, stream>>>(bufAgg, bufMean, bufVar);
            bn_relu_kernel<<<(tot + 255) / 256, 256, 0, stream>>>(bufAgg, bufMean, bufVar,
                                                                  g[L], bb[L], bufX);
            x = bufX;
        } else {
            head_mean_kernel<<<((N_NODES * 40) + 255) / 256, 256, 0, stream>>>(
                bufAgg, bias_last, (float*)d_out);
        }
    }
}